// MoVE_12386685682260
// MI455X (gfx1250) — compile-verified
//
#include <hip/hip_runtime.h>
#include <hip/hip_bf16.h>
#include <math.h>

// ---------------------------------------------------------------------------
// MoVE pipeline for MI455X (gfx1250, wave32).
//   k1: gate (3x3 pool -> MLP -> sigmoid)          [B,C,A]
//   k2: depthwise conv stats (sum/sumsq per b,c,a) -> InstanceNorm mean/rstd
//   k3: depthwise conv recompute + IN + SiLU + gated sum over experts -> moe
//   k4: 1x1 projection GEMM via V_WMMA_F32_16X16X4_F32  -> z (in d_out)
//   k5: BatchNorm stats per output channel
//   k6: BN affine + SiLU, in place on d_out
// ---------------------------------------------------------------------------

#define BB   4
#define CC   256
#define OO   256
#define AA   9
#define HH   96
#define WW   96
#define HWQ  (HH * WW)          // 9216
#define HIDN 18
#define EPSI 1e-5f

typedef __attribute__((ext_vector_type(2))) float f32x2;
typedef __attribute__((ext_vector_type(8))) float f32x8;

__device__ __forceinline__ float dsigmoid(float v) { return 1.0f / (1.0f + __expf(-v)); }
__device__ __forceinline__ float dsilu(float v)    { return v * dsigmoid(v); }

__device__ __forceinline__ float wave_sum32(float v) {
#pragma unroll
  for (int off = 16; off > 0; off >>= 1) v += __shfl_xor(v, off, 32);
  return v;
}

// ---------------------------------------------------------------- k1: gate --
// grid = B*C, block = 288 (9 waves, one per pooled cell)
__global__ __launch_bounds__(288) void k1_gate(
    const float* __restrict__ x, const float* __restrict__ w_g1,
    const float* __restrict__ b_g1, const float* __restrict__ w_g2,
    const float* __restrict__ b_g2, float* __restrict__ gw) {
  __shared__ float sPool[AA];
  __shared__ float sHid[HIDN];

  const int bc   = blockIdx.x;                 // (b*C + c)
  const int wid  = threadIdx.x >> 5;           // 0..8 = pooled cell a
  const int lane = threadIdx.x & 31;

  const float* xc = x + (size_t)bc * HWQ;
  const int rblk = wid / 3, sblk = wid % 3;
  const int row  = rblk * 32 + lane;
  const float* rp = xc + row * WW + sblk * 32;
  float s = 0.0f;
#pragma unroll 8
  for (int j = 0; j < 32; ++j) s += rp[j];
  s = wave_sum32(s);
  if (lane == 0) sPool[wid] = s * (1.0f / 1024.0f);
  __syncthreads();

  const int t = threadIdx.x;
  if (t < HIDN) {
    float h = b_g1[t];
#pragma unroll
    for (int a = 0; a < AA; ++a) h += sPool[a] * w_g1[t * AA + a];
    sHid[t] = dsilu(h);
  }
  __syncthreads();
  if (t < AA) {
    float v = b_g2[t];
#pragma unroll
    for (int hh = 0; hh < HIDN; ++hh) v += sHid[hh] * w_g2[t * HIDN + hh];
    gw[(size_t)bc * AA + t] = dsigmoid(v);
  }
}

// --------------------------- shared conv helper: 3x3 taps from LDS channel --
__device__ __forceinline__ void load_taps(const float* sX, int p, float nb[9]) {
  const int h = p / WW, w = p % WW;
#pragma unroll
  for (int dy = 0; dy < 3; ++dy)
#pragma unroll
    for (int dx = 0; dx < 3; ++dx) {
      const int hh = h + dy - 1, ww = w + dx - 1;
      nb[dy * 3 + dx] =
          (hh >= 0 && hh < HH && ww >= 0 && ww < WW) ? sX[hh * WW + ww] : 0.0f;
    }
}

// ----------------------------------------------- k2: InstanceNorm statistics
// grid = B*C, block = 256
__global__ __launch_bounds__(256) void k2_instats(
    const float* __restrict__ x, const float* __restrict__ w_expert,
    float* __restrict__ iMean, float* __restrict__ iRstd) {
  __shared__ float sX[HWQ];
  __shared__ float sWk[AA * 9];
  __shared__ float sAcc[2 * AA];

  const int bc = blockIdx.x;
  const int c  = bc % CC;
  const int tid = threadIdx.x;

  for (int i = tid; i < HWQ; i += 256) sX[i] = x[(size_t)bc * HWQ + i];
  if (tid < AA * 9) sWk[tid] = w_expert[(size_t)c * (AA * 9) + tid];
  if (tid < 2 * AA) sAcc[tid] = 0.0f;
  __syncthreads();

  float as[AA], aq[AA];
#pragma unroll
  for (int a = 0; a < AA; ++a) { as[a] = 0.0f; aq[a] = 0.0f; }

  for (int p = tid; p < HWQ; p += 256) {
    float nb[9];
    load_taps(sX, p, nb);
#pragma unroll
    for (int a = 0; a < AA; ++a) {
      float v = 0.0f;
#pragma unroll
      for (int j = 0; j < 9; ++j) v = fmaf(nb[j], sWk[a * 9 + j], v);
      as[a] += v;
      aq[a] = fmaf(v, v, aq[a]);
    }
  }
  const int lane = tid & 31;
#pragma unroll
  for (int a = 0; a < AA; ++a) {
    float s = wave_sum32(as[a]);
    float q = wave_sum32(aq[a]);
    if (lane == 0) { atomicAdd(&sAcc[a], s); atomicAdd(&sAcc[AA + a], q); }
  }
  __syncthreads();
  if (tid < AA) {
    const float m = sAcc[tid] * (1.0f / (float)HWQ);
    const float q = sAcc[AA + tid] * (1.0f / (float)HWQ);
    iMean[(size_t)bc * AA + tid] = m;
    iRstd[(size_t)bc * AA + tid] = rsqrtf(q - m * m + EPSI);
  }
}

// ------------------------------- k3: conv + IN + SiLU + gated sum -> moe ----
// grid = B*C, block = 256
__global__ __launch_bounds__(256) void k3_moe(
    const float* __restrict__ x, const float* __restrict__ w_expert,
    const float* __restrict__ gw, const float* __restrict__ iMean,
    const float* __restrict__ iRstd, float* __restrict__ moe) {
  __shared__ float sX[HWQ];
  __shared__ float sWk[AA * 9];
  __shared__ float sM[AA], sR[AA], sG[AA];

  const int bc = blockIdx.x;
  const int c  = bc % CC;
  const int tid = threadIdx.x;

  for (int i = tid; i < HWQ; i += 256) sX[i] = x[(size_t)bc * HWQ + i];
  if (tid < AA * 9) sWk[tid] = w_expert[(size_t)c * (AA * 9) + tid];
  if (tid < AA) {
    sM[tid] = iMean[(size_t)bc * AA + tid];
    sR[tid] = iRstd[(size_t)bc * AA + tid];
    sG[tid] = gw[(size_t)bc * AA + tid];
  }
  __syncthreads();

  for (int p = tid; p < HWQ; p += 256) {
    float nb[9];
    load_taps(sX, p, nb);
    float acc = 0.0f;
#pragma unroll
    for (int a = 0; a < AA; ++a) {
      float v = 0.0f;
#pragma unroll
      for (int j = 0; j < 9; ++j) v = fmaf(nb[j], sWk[a * 9 + j], v);
      const float vn = (v - sM[a]) * sR[a];
      acc = fmaf(sG[a], dsilu(vn), acc);
    }
    moe[(size_t)bc * HWQ + p] = acc;
  }
}

// ------------------- k4: 1x1 projection GEMM, V_WMMA_F32_16X16X4_F32 -------
// grid = (HW/128, O/16, B), block = 256 (8 waves; wave -> one 16x16 tile)
#define SWSTR 260  // padded LDS row stride (banks spread: 260 % 64 == 4)
__global__ __launch_bounds__(256) void k4_proj_wmma(
    const float* __restrict__ moe, const float* __restrict__ wproj,
    float* __restrict__ z) {
  __shared__ float sW[16 * SWSTR];

  const int oBase = blockIdx.y * 16;
  const int b     = blockIdx.z;

  // stage 16 rows of w_proj (row-major [O, C]) into LDS
  for (int i = threadIdx.x; i < 16 * CC; i += 256) {
    const int r = i >> 8, k = i & 255;
    sW[r * SWSTR + k] = wproj[(size_t)(oBase + r) * CC + k];
  }
  __syncthreads();

  const int wid  = threadIdx.x >> 5;
  const int lane = threadIdx.x & 31;
  const int half = lane >> 4;    // K sub-pair selector
  const int mn   = lane & 15;    // M for A-frag, N for B-frag / D cols

  const int pixBase = blockIdx.x * 128 + wid * 16;
  const float* mb = moe + (size_t)b * CC * HWQ + pixBase + mn;

  f32x8 acc = {};
  for (int k0 = 0; k0 < CC; k0 += 4) {
    const int kk = k0 + 2 * half;
    f32x2 afr, bfr;
    afr.x = sW[mn * SWSTR + kk];
    afr.y = sW[mn * SWSTR + kk + 1];
    const float* p = mb + (size_t)kk * HWQ;
    bfr.x = p[0];
    bfr.y = p[HWQ];
    if (k0 < CC - 16) __builtin_prefetch(p + (size_t)4 * HWQ, 0, 1);
    // D = A(16x4 f32) * B(4x16 f32) + C : full fp32 precision
    acc = __builtin_amdgcn_wmma_f32_16x16x4_f32(
        /*neg_a=*/false, afr, /*neg_b=*/false, bfr,
        /*c_mod=*/(short)0, acc, /*reuse_a=*/false, /*reuse_b=*/false);
  }

  union { f32x8 v; float f[8]; } u;
  u.v = acc;
  float* zb = z + ((size_t)b * OO + oBase + half * 8) * HWQ + pixBase + mn;
#pragma unroll
  for (int r = 0; r < 8; ++r) zb[(size_t)r * HWQ] = u.f[r];
}

// ------------------------------------------ k5: BatchNorm stats per channel
// grid = O, block = 256
__global__ __launch_bounds__(256) void k5_bnstats(
    const float* __restrict__ z, float* __restrict__ bnM,
    float* __restrict__ bnR) {
  __shared__ float rs[256], rq[256];
  const int o = blockIdx.x, tid = threadIdx.x;
  float s = 0.0f, q = 0.0f;
  for (int b = 0; b < BB; ++b) {
    const float* p = z + ((size_t)b * OO + o) * HWQ;
    for (int i = tid; i < HWQ; i += 256) {
      const float v = p[i];
      s += v;
      q = fmaf(v, v, q);
    }
  }
  rs[tid] = s; rq[tid] = q;
  __syncthreads();
  for (int off = 128; off > 0; off >>= 1) {
    if (tid < off) { rs[tid] += rs[tid + off]; rq[tid] += rq[tid + off]; }
    __syncthreads();
  }
  if (tid == 0) {
    const float inv = 1.0f / (float)(BB * HWQ);
    const float m = rs[0] * inv;
    bnM[o] = m;
    bnR[o] = rsqrtf(rq[0] * inv - m * m + EPSI);
  }
}

// -------------------------------------- k6: BN affine + SiLU (in place) ----
__global__ __launch_bounds__(256) void k6_bnapply(
    float* __restrict__ z, const float* __restrict__ bnM,
    const float* __restrict__ bnR, const float* __restrict__ gamma,
    const float* __restrict__ beta) {
  const size_t total = (size_t)BB * OO * HWQ;
  for (size_t idx = (size_t)blockIdx.x * blockDim.x + threadIdx.x; idx < total;
       idx += (size_t)gridDim.x * blockDim.x) {
    const int o = (int)((idx / HWQ) % OO);
    const float t = (z[idx] - bnM[o]) * bnR[o] * gamma[o] + beta[o];
    z[idx] = dsilu(t);
  }
}

// ---------------------------------------------------------------------------
extern "C" void kernel_launch(void* const* d_in, const int* in_sizes, int n_in,
                              void* d_out, int out_size, void* d_ws,
                              size_t ws_size, hipStream_t stream) {
  const float* x        = (const float*)d_in[0];
  const float* w_expert = (const float*)d_in[1];
  const float* w_g1     = (const float*)d_in[2];
  const float* b_g1     = (const float*)d_in[3];
  const float* w_g2     = (const float*)d_in[4];
  const float* b_g2     = (const float*)d_in[5];
  const float* w_proj   = (const float*)d_in[6];
  const float* bn_gamma = (const float*)d_in[7];
  const float* bn_beta  = (const float*)d_in[8];
  float* z = (float*)d_out;  // [B,O,H,W] — GEMM result, finalized in place

  float* ws     = (float*)d_ws;
  float* gw     = ws;                       // B*C*A = 9216
  float* iMean  = gw + (size_t)BB * CC * AA;
  float* iRstd  = iMean + (size_t)BB * CC * AA;
  float* bnM    = iRstd + (size_t)BB * CC * AA;
  float* bnR    = bnM + OO;
  float* moe    = bnR + OO;                 // B*C*H*W = 9.44M floats

  k1_gate<<<BB * CC, 288, 0, stream>>>(x, w_g1, b_g1, w_g2, b_g2, gw);
  k2_instats<<<BB * CC, 256, 0, stream>>>(x, w_expert, iMean, iRstd);
  k3_moe<<<BB * CC, 256, 0, stream>>>(x, w_expert, gw, iMean, iRstd, moe);
  dim3 g4(HWQ / 128, OO / 16, BB);
  k4_proj_wmma<<<g4, 256, 0, stream>>>(moe, w_proj, z);
  k5_bnstats<<<OO, 256, 0, stream>>>(z, bnM, bnR);
  k6_bnapply<<<4096, 256, 0, stream>>>(z, bnM, bnR, bn_gamma, bn_beta);
}